// SSIMLoss_64750926954835
// MI455X (gfx1250) — compile-verified
//
#include <hip/hip_runtime.h>

#define WIN 7
#define IMG 512
#define OUTD 506                 // 512 - 7 + 1
#define NIMG 80                  // 8 * 10
#define TILES_PER_SIDE 32        // ceil(506/16)
#define TILES_PER_IMG (TILES_PER_SIDE * TILES_PER_SIDE)   // 1024
#define NTILES (NIMG * TILES_PER_IMG)                     // 81920
#define WAVES 4

typedef __attribute__((ext_vector_type(2))) float v2f;
typedef __attribute__((ext_vector_type(8))) float v8f;

// ---------------- Kernel 1: per-image dynamic range ----------------
__global__ __launch_bounds__(256) void ssim_range_kernel(
    const float* __restrict__ X, const float* __restrict__ Y,
    float* __restrict__ range) {
  const int img = blockIdx.x;
  const float4* Xi = (const float4*)(X + (size_t)img * IMG * IMG);
  const float4* Yi = (const float4*)(Y + (size_t)img * IMG * IMG);
  float mnx = 1e30f, mxx = -1e30f, mny = 1e30f, mxy = -1e30f;
  for (int i = threadIdx.x; i < IMG * IMG / 4; i += 256) {
    float4 a = Xi[i];
    mnx = fminf(mnx, fminf(fminf(a.x, a.y), fminf(a.z, a.w)));
    mxx = fmaxf(mxx, fmaxf(fmaxf(a.x, a.y), fmaxf(a.z, a.w)));
    float4 b = Yi[i];
    mny = fminf(mny, fminf(fminf(b.x, b.y), fminf(b.z, b.w)));
    mxy = fmaxf(mxy, fmaxf(fmaxf(b.x, b.y), fmaxf(b.z, b.w)));
  }
  #pragma unroll
  for (int off = 16; off > 0; off >>= 1) {
    mnx = fminf(mnx, __shfl_xor(mnx, off, 32));
    mxx = fmaxf(mxx, __shfl_xor(mxx, off, 32));
    mny = fminf(mny, __shfl_xor(mny, off, 32));
    mxy = fmaxf(mxy, __shfl_xor(mxy, off, 32));
  }
  __shared__ float s[8][4];
  const int wid = threadIdx.x >> 5, lane = threadIdx.x & 31;
  if (lane == 0) { s[wid][0] = mnx; s[wid][1] = mxx; s[wid][2] = mny; s[wid][3] = mxy; }
  __syncthreads();
  if (threadIdx.x == 0) {
    float a = s[0][0], b = s[0][1], c = s[0][2], d = s[0][3];
    #pragma unroll
    for (int w = 1; w < 8; ++w) {
      a = fminf(a, s[w][0]); b = fmaxf(b, s[w][1]);
      c = fminf(c, s[w][2]); d = fmaxf(d, s[w][3]);
    }
    range[img] = fmaxf(b - a, d - c);
  }
}

// ---------------- Kernel 2: SSIM tiles (WMMA horizontal box sum) ----------------
// One wave per 16x16 output tile. Lane t owns image column c0+t (22 rows halo):
// vertical 7-sum via register prefix -> LDS V[q][16][24]; horizontal 7-sum as
// out(16x16) = V(16x24) x BandMatrix(24x16) using 6x v_wmma_f32_16x16x4_f32.
__global__ __launch_bounds__(32 * WAVES) void ssim_tile_kernel(
    const float* __restrict__ X, const float* __restrict__ Y,
    const float* __restrict__ range, float* __restrict__ partial) {
  __shared__ float V[WAVES][5][16][24];
  const int wid  = threadIdx.x >> 5;
  const int lane = threadIdx.x & 31;
  const int tile = blockIdx.x * WAVES + wid;
  const int img  = tile >> 10;
  const int tin  = tile & 1023;
  const int r0   = (tin >> 5) << 4;
  const int c0   = (tin & 31) << 4;
  const float* Xi = X + (size_t)img * IMG * IMG;
  const float* Yi = Y + (size_t)img * IMG * IMG;

  float cx[22], cy[22];
  if (lane < 22) {
    const int col = min(c0 + lane, IMG - 1);
    #pragma unroll
    for (int r = 0; r < 22; ++r) {
      const int rr = min(r0 + r, IMG - 1);
      cx[r] = Xi[rr * IMG + col];
      cy[r] = Yi[rr * IMG + col];
    }
  } else {
    #pragma unroll
    for (int r = 0; r < 22; ++r) { cx[r] = 0.f; cy[r] = 0.f; }
  }

  float (*Vw)[16][24] = V[wid];
  #pragma unroll
  for (int q = 0; q < 5; ++q) {
    float buf[23];
    buf[0] = 0.f;
    #pragma unroll
    for (int r = 0; r < 22; ++r) {
      float val;
      if      (q == 0) val = cx[r];
      else if (q == 1) val = cy[r];
      else if (q == 2) val = cx[r] * cx[r];
      else if (q == 3) val = cy[r] * cy[r];
      else             val = cx[r] * cy[r];
      buf[r + 1] = buf[r] + val;
    }
    if (lane < 24) {   // lanes 22,23 write zeros into K-pad columns (B rows there are 0 anyway)
      #pragma unroll
      for (int i = 0; i < 16; ++i)
        Vw[q][i][lane] = (lane < 22) ? (buf[i + 7] - buf[i]) : 0.f;
    }
  }
  __syncthreads();

  // WMMA fragments: A f32 16x4 layout -> lanes 0-15: M=lane, K=k0,k0+1 ; lanes 16-31: M=lane-16, K=k0+2,k0+3
  const int half = lane >> 4;
  const int m    = lane & 15;   // M for A, N for B/C/D
  v8f acc[5];
  #pragma unroll
  for (int q = 0; q < 5; ++q) acc[q] = (v8f){0.f,0.f,0.f,0.f,0.f,0.f,0.f,0.f};

  #pragma unroll
  for (int kb = 0; kb < 6; ++kb) {
    const int k0 = kb * 4 + half * 2;
    v2f bfrag;  // band matrix S[k][n] = (n <= k <= n+6)
    bfrag.x = (k0     >= m && k0     <= m + 6) ? 1.f : 0.f;
    bfrag.y = (k0 + 1 >= m && k0 + 1 <= m + 6) ? 1.f : 0.f;
    #pragma unroll
    for (int q = 0; q < 5; ++q) {
      v2f afrag;
      afrag.x = Vw[q][m][k0];
      afrag.y = Vw[q][m][k0 + 1];
      acc[q] = __builtin_amdgcn_wmma_f32_16x16x4_f32(
          false, afrag, false, bfrag, (short)0, acc[q], false, false);
    }
  }

  // Per-element SSIM.  D layout: VGPR v -> M = v + 8*half, N = lane&15
  const float rg = range[img];
  float C1 = 0.01f * rg; C1 *= C1;
  float C2 = 0.03f * rg; C2 *= C2;
  const float inv49 = 1.f / 49.f;
  const float covn  = 49.f / 48.f;
  float ssum = 0.f;
  #pragma unroll
  for (int v = 0; v < 8; ++v) {
    const int orow = r0 + v + 8 * half;
    const int ocol = c0 + m;
    const float sx = acc[0][v], sy = acc[1][v];
    const float sxx = acc[2][v], syy = acc[3][v], sxy = acc[4][v];
    const float ux = sx * inv49, uy = sy * inv49;
    const float vx  = covn * (sxx * inv49 - ux * ux);
    const float vy  = covn * (syy * inv49 - uy * uy);
    const float vxy = covn * (sxy * inv49 - ux * uy);
    const float A1 = 2.f * ux * uy + C1;
    const float A2 = 2.f * vxy + C2;
    const float B1 = ux * ux + uy * uy + C1;
    const float B2 = vx + vy + C2;
    const float S = (A1 * A2) / (B1 * B2);
    if (orow < OUTD && ocol < OUTD) ssum += S;
  }
  #pragma unroll
  for (int off = 16; off > 0; off >>= 1) ssum += __shfl_xor(ssum, off, 32);
  if (lane == 0) partial[tile] = ssum;
}

// ---------------- Kernel 3: final reduction ----------------
__global__ __launch_bounds__(1024) void ssim_final_kernel(
    const float* __restrict__ partial, float* __restrict__ out) {
  __shared__ double s[1024];
  double acc = 0.0;
  for (int i = threadIdx.x; i < NTILES; i += 1024) acc += (double)partial[i];
  s[threadIdx.x] = acc;
  __syncthreads();
  #pragma unroll
  for (int off = 512; off > 0; off >>= 1) {
    if (threadIdx.x < off) s[threadIdx.x] += s[threadIdx.x + off];
    __syncthreads();
  }
  if (threadIdx.x == 0) {
    const double cnt = (double)NIMG * (double)OUTD * (double)OUTD;
    out[0] = 1.f - (float)(s[0] / cnt);
  }
}

extern "C" void kernel_launch(void* const* d_in, const int* in_sizes, int n_in,
                              void* d_out, int out_size, void* d_ws, size_t ws_size,
                              hipStream_t stream) {
  const float* X = (const float*)d_in[0];
  const float* Y = (const float*)d_in[1];
  float* out     = (float*)d_out;
  float* range   = (float*)d_ws;                        // 80 floats
  float* partial = (float*)((char*)d_ws + 512);         // NTILES floats

  ssim_range_kernel<<<NIMG, 256, 0, stream>>>(X, Y, range);
  ssim_tile_kernel<<<NTILES / WAVES, 32 * WAVES, 0, stream>>>(X, Y, range, partial);
  ssim_final_kernel<<<1, 1024, 0, stream>>>(partial, out);
}